// WaveletTransform3D_24300924961070
// MI455X (gfx1250) — compile-verified
//
#include <hip/hip_runtime.h>
#include <stdint.h>

// 3D Haar DWT, stride-2 kernel-2 => non-overlapping 2x2x2 blocks.
// Input : [B=8, C=16, T=16, H=256, W=256] f32   (512 MiB)
// Output: 8 subbands, each [8,16,8,128,128] f32, concatenated (512 MiB)
// Pure streaming problem: ~1 GiB traffic -> HBM-bound (~46us @ 23.3 TB/s).
// Uses CDNA5 async global->LDS B128 loads (ASYNCcnt) for staging.

#define DIM_T 16
#define DIM_H 256
#define DIM_W 256
// bc = b*16+c in [0,128); outputs: to in [0,8), ho in [0,128), wo in [0,128)

typedef float v4f __attribute__((ext_vector_type(4)));

__device__ __forceinline__ v4f f4make(float a, float b, float c, float d) {
    v4f r; r.x = a; r.y = b; r.z = c; r.w = d; return r;
}

// CDNA5 async tensor-path load: 16B global -> LDS, tracked by ASYNCcnt.
__device__ __forceinline__ void async_load_b128(unsigned lds_addr, const float* gptr) {
    unsigned long long ga = (unsigned long long)(uintptr_t)gptr;
    asm volatile("global_load_async_to_lds_b128 %0, %1, off"
                 :: "v"(lds_addr), "v"(ga)
                 : "memory");
}
__device__ __forceinline__ void wait_asynccnt0() {
    asm volatile("s_wait_asynccnt 0x0" ::: "memory");
}

__global__ __launch_bounds__(256) void haar3d_fused_kernel(
    const float* __restrict__ in, float* __restrict__ out)
{
    // Plane-major LDS staging: plane j holds one B128 per thread.
    // ds_load_b128 of a plane is bank-conflict-free (lanes hit consecutive 16B).
    __shared__ v4f buf[8][256];   // 32 KiB

    const int tid = threadIdx.x;
    const int g   = blockIdx.x * 256 + tid;

    const int wv = g & 31;           // 32 groups of 4 wo each (wo = 4*wv..4*wv+3)
    const int ho = (g >> 5) & 127;
    const int to = (g >> 12) & 7;
    const int bc = g >> 15;          // 0..127

    const int t0 = 2 * to;
    const int h0 = 2 * ho;
    const int w0 = 8 * wv;

    // Issue 8 async B128 loads (rows (t,h) in {0,1}^2, two 16B halves along w).
    #pragma unroll
    for (int r = 0; r < 4; ++r) {
        const int t = t0 + (r >> 1);
        const int h = h0 + (r & 1);
        const float* gp = in + (((size_t)(bc * DIM_T + t) * DIM_H + h) * DIM_W + w0);
        #pragma unroll
        for (int j = 0; j < 2; ++j) {
            const unsigned lds_addr =
                (unsigned)(uintptr_t)(&buf[r * 2 + j][tid]);   // low 32 bits = LDS offset
            async_load_b128(lds_addr, gp + 4 * j);
        }
    }
    wait_asynccnt0();

    v4f A[4][2];
    #pragma unroll
    for (int r = 0; r < 4; ++r) {
        A[r][0] = buf[r * 2 + 0][tid];
        A[r][1] = buf[r * 2 + 1][tid];
    }

    // --- w stage: pairwise (x[2k] +/- x[2k+1]) within each row -> 4 wo lanes ---
    v4f Wp[4], Wm[4];
    #pragma unroll
    for (int r = 0; r < 4; ++r) {
        const v4f a = A[r][0];
        const v4f b = A[r][1];
        Wp[r] = f4make(a.x + a.y, a.z + a.w, b.x + b.y, b.z + b.w);
        Wm[r] = f4make(a.x - a.y, a.z - a.w, b.x - b.y, b.z - b.w);
    }

    // --- h stage: rows r = 2*t + h ---
    v4f hPp[2], hMp[2], hPm[2], hMm[2];
    #pragma unroll
    for (int t = 0; t < 2; ++t) {
        hPp[t] = Wp[2 * t] + Wp[2 * t + 1];   // h-lo of w-lo
        hMp[t] = Wp[2 * t] - Wp[2 * t + 1];   // h-hi of w-lo
        hPm[t] = Wm[2 * t] + Wm[2 * t + 1];   // h-lo of w-hi
        hMm[t] = Wm[2 * t] - Wm[2 * t + 1];   // h-hi of w-hi
    }

    // --- t stage + scale; subband s = ft*4 + fh*2 + fw (LLL..HHH) ---
    const float S3 = 0.35355339059327373f;    // (1/sqrt(2))^3
    v4f o[8];
    o[0] = (hPp[0] + hPp[1]) * S3;            // LLL
    o[1] = (hPm[0] + hPm[1]) * S3;            // LLH
    o[2] = (hMp[0] + hMp[1]) * S3;            // LHL
    o[3] = (hMm[0] + hMm[1]) * S3;            // LHH
    o[4] = (hPp[0] - hPp[1]) * S3;            // HLL
    o[5] = (hPm[0] - hPm[1]) * S3;            // HLH
    o[6] = (hMp[0] - hMp[1]) * S3;            // HHL
    o[7] = (hMm[0] - hMm[1]) * S3;            // HHH

    // Streaming stores: subband s at offset s*Nsub, fully coalesced 16B/lane.
    const size_t Nsub = (size_t)128 * 8 * 128 * 128;          // 16,777,216
    float* op = out + (((size_t)(bc * 8 + to) * 128 + ho) * 128 + 4 * wv);
    #pragma unroll
    for (int s = 0; s < 8; ++s) {
        __builtin_nontemporal_store(o[s], (v4f*)(op + (size_t)s * Nsub));
    }
}

extern "C" void kernel_launch(void* const* d_in, const int* in_sizes, int n_in,
                              void* d_out, int out_size, void* d_ws, size_t ws_size,
                              hipStream_t stream) {
    (void)in_sizes; (void)n_in; (void)out_size; (void)d_ws; (void)ws_size;
    const float* in = (const float*)d_in[0];
    float* out = (float*)d_out;
    // 128 (bc) * 8 (to) * 128 (ho) * 32 (wv) threads = 4,194,304 = 16384 blocks x 256
    haar3d_fused_kernel<<<dim3(16384), dim3(256), 0, stream>>>(in, out);
}